// VGGPerceptual_37237366456609
// MI455X (gfx1250) — compile-verified
//
#include <hip/hip_runtime.h>
#include <hip/hip_bf16.h>
#include <math.h>

typedef float v2f __attribute__((ext_vector_type(2)));
typedef float v8f __attribute__((ext_vector_type(8)));

// Numerically-safe merge of two (max, sumexp) online-logsumexp states.
// Guards the m == nm case so (-inf) - (-inf) never produces NaN.
__device__ __forceinline__ void lse_merge(float& m, float& s, float om, float os) {
    float nm = fmaxf(m, om);
    float ea = (m == nm) ? 1.0f : __expf(m - nm);
    float eb = (om == nm) ? 1.0f : __expf(om - nm);
    s = s * ea + os * eb;
    m = nm;
}

// One workgroup (256 threads = 8 wave32) per row. Streams the row once with
// 16B-aligned float4 loads, keeps an online (max, sum-exp) per thread,
// reduces wave -> block, writes item_loss[row] = logsumexp(row) - row[target].
__global__ void __launch_bounds__(256)
ce_row_kernel(const float* __restrict__ inp,
              const int* __restrict__ tgt,
              float* __restrict__ item_loss,
              int V)
{
    const int row = blockIdx.x;
    const int tid = threadIdx.x;
    const float* __restrict__ rp = inp + (size_t)row * (size_t)V;

    // Align vector region to 16 bytes: global element index of row start is
    // row*V; prologue of p (<4) scalars brings us to a float4 boundary.
    const int a  = (int)(((size_t)row * (size_t)V) & 3);
    const int p  = (4 - a) & 3;
    const int Nv = (V - p) >> 2;          // number of float4s
    const int e0 = p + (Nv << 2);         // epilogue start
    const int rem = V - e0;               // 0..3 scalars

    float m = -INFINITY;
    float s = 0.0f;

    const float4* __restrict__ vp = reinterpret_cast<const float4*>(rp + p);
    for (int i = tid; i < Nv; i += 256) {
        float4 v = vp[i];
        float mx = fmaxf(fmaxf(v.x, v.y), fmaxf(v.z, v.w));
        float s4 = __expf(v.x - mx) + __expf(v.y - mx) +
                   __expf(v.z - mx) + __expf(v.w - mx);
        // merge (mx, s4) into running (m, s); first iter has s==0 so the
        // 0 * exp(-inf) = 0 path is benign.
        float nm = fmaxf(m, mx);
        s = s * __expf(m - nm) + s4 * __expf(mx - nm);
        m = nm;
    }

    // Ragged edges (at most 3 + 3 scalars per row, handled by low threads).
    if (tid < p)   lse_merge(m, s, rp[tid], 1.0f);
    if (tid < rem) lse_merge(m, s, rp[e0 + tid], 1.0f);

    // Wave32 reduction.
    #pragma unroll
    for (int off = 16; off > 0; off >>= 1) {
        float om = __shfl_xor(m, off, 32);
        float os = __shfl_xor(s, off, 32);
        lse_merge(m, s, om, os);
    }

    __shared__ float sm[8];
    __shared__ float ss[8];
    const int wave = tid >> 5;
    const int lane = tid & 31;
    if (lane == 0) { sm[wave] = m; ss[wave] = s; }
    __syncthreads();

    if (tid == 0) {
        #pragma unroll
        for (int w = 1; w < 8; ++w) lse_merge(m, s, sm[w], ss[w]);
        const float lse    = m + logf(s);
        const float picked = rp[tgt[row]];
        item_loss[row] = lse - picked;
    }
}

// Final mean over n per-row losses, computed as ones^T * x on the tensor core:
// V_WMMA_F32_16X16X4_F32 with B = all-ones. A-tile layout per CDNA5 ISA
// 7.12.2 (32-bit A 16x4): lanes 0-15 hold (M=lane, K=0..1) in VGPR0..1,
// lanes 16-31 hold (M=lane-16, K=2..3). With B==1, every D column holds the
// 16 row-sums, so total = sum_m D[m,0] = lanesum(lane 0) + lanesum(lane 16).
__global__ void __launch_bounds__(32)
ce_reduce_wmma(const float* __restrict__ x,
               float* __restrict__ out,
               int n, float scale)
{
    const int lane = threadIdx.x;            // single wave32, EXEC all ones
    const int n64  = n & ~63;

#if __has_builtin(__builtin_amdgcn_wmma_f32_16x16x4_f32)
    v8f c = {};
    v2f b; b[0] = 1.0f; b[1] = 1.0f;
    const int mrow = lane & 15;
    const int k0   = (lane >> 4) << 1;
    for (int base = 0; base < n64; base += 64) {
        v2f av;
        av[0] = x[base + mrow * 4 + k0];
        av[1] = x[base + mrow * 4 + k0 + 1];
        // D = A * ones + C  (f32 accumulate, exact row-sum chaining)
        c = __builtin_amdgcn_wmma_f32_16x16x4_f32(
                /*neg_a=*/false, av, /*neg_b=*/false, b,
                /*c_mod=*/(short)0, c, /*reuse_a=*/false, /*reuse_b=*/false);
    }
    float lsum = c[0] + c[1] + c[2] + c[3] + c[4] + c[5] + c[6] + c[7];
#else
    // Fallback: plain strided sum of the 64-aligned region.
    float lsum_acc = 0.0f;
    for (int i = lane; i < n64; i += 32) lsum_acc += x[i];
    float lsum = (lane == 0 || lane == 16) ? 0.0f : 0.0f;
    // fold strided sum into the lane0/lane16 convention via full reduce below
    float full = lsum_acc;
    #pragma unroll
    for (int off = 16; off > 0; off >>= 1) full += __shfl_xor(full, off, 32);
    lsum = (lane == 0) ? full : 0.0f;   // lane16 contributes 0
#endif

    // Tail (n not multiple of 64): classic shuffle reduction.
    float tail = 0.0f;
    for (int i = n64 + lane; i < n; i += 32) tail += x[i];
    #pragma unroll
    for (int off = 16; off > 0; off >>= 1) tail += __shfl_xor(tail, off, 32);

    const float tot = __shfl(lsum, 0, 32) + __shfl(lsum, 16, 32) + tail;
    if (lane == 0) out[0] = tot * scale;
}

extern "C" void kernel_launch(void* const* d_in, const int* in_sizes, int n_in,
                              void* d_out, int out_size, void* d_ws, size_t ws_size,
                              hipStream_t stream)
{
    const float* inp = (const float*)d_in[0];   // [B, V] f32 logits
    const int*   tgt = (const int*)d_in[1];     // [B] targets
    float*       out = (float*)d_out;           // scalar f32
    float*       ws  = (float*)d_ws;            // B floats of per-row loss

    const int B = in_sizes[1];
    const int V = in_sizes[0] / B;

    ce_row_kernel<<<B, 256, 0, stream>>>(inp, tgt, ws, V);
    ce_reduce_wmma<<<1, 32, 0, stream>>>(ws, out, B, 1.0f / (float)B);
}